// PyKeOpsSimilarity_80135499809316
// MI455X (gfx1250) — compile-verified
//
#include <hip/hip_runtime.h>

typedef __attribute__((ext_vector_type(2))) float v2f;
typedef __attribute__((ext_vector_type(8))) float v8f;

#define BATCH   4
#define NX      4096
#define NY      8192
#define DIM     256
#define TOPK    10
#define CHUNK   512
#define SW      544            // scan width: CHUNK + 10 appended + 22 pad (multiple of 32)
#define INV_TAU 20.0f
#define TOTN    (BATCH * NX * TOPK)
#define NEG_INF (-__builtin_inff())

// ---------------------------------------------------------------------------
// Kernel 1: per-row inverse L2 norms: inv = 1 / max(||v||, 1e-12)
// One wave (32 lanes) per row; 256 floats -> 8 per lane; wave32 shfl reduce.
// ---------------------------------------------------------------------------
__global__ __launch_bounds__(256) void norms_kernel(const float* __restrict__ X,
                                                    const float* __restrict__ Y,
                                                    float* __restrict__ invx,
                                                    float* __restrict__ invy) {
  int gw   = (blockIdx.x * 256 + threadIdx.x) >> 5;   // global wave id = row id
  int lane = threadIdx.x & 31;
  const float* src;
  float* dst;
  if (gw < BATCH * NX) {
    src = X + (size_t)gw * DIM;
    dst = invx + gw;
  } else {
    int r = gw - BATCH * NX;
    src = Y + (size_t)r * DIM;
    dst = invy + r;
  }
  float s = 0.f;
#pragma unroll
  for (int j = 0; j < DIM / 32; ++j) {
    float v = src[lane + 32 * j];
    s += v * v;
  }
#pragma unroll
  for (int off = 16; off; off >>= 1) s += __shfl_xor(s, off, 32);
  if (lane == 0) {
    float n = fmaxf(sqrtf(s), 1e-12f);
    *dst = 1.0f / n;
  }
}

// ---------------------------------------------------------------------------
// Kernel 2: fused normalized GEMM (f32 WMMA) + streaming per-row top-10
//           + softmax + sort-by-column + sparse COO emission.
// Block = (b, 16-row tile), 256 threads = 8 waves.
// ---------------------------------------------------------------------------
__global__ __launch_bounds__(256) void sim_topk_kernel(const float* __restrict__ X,
                                                       const float* __restrict__ Y,
                                                       const float* __restrict__ invx,
                                                       const float* __restrict__ invy,
                                                       float* __restrict__ outV,
                                                       int* __restrict__ outI) {
  __shared__ float simS[16 * SW];       // chunk of scaled sims (+ appended running top)
  __shared__ float topvS[16][TOPK];     // running top-10 values (desc)
  __shared__ int   topcS[16][TOPK];     // running top-10 columns
  __shared__ float ntvS[16][TOPK];      // new top scratch
  __shared__ int   ntcS[16][TOPK];
  __shared__ float invxS[16];

  const int b    = blockIdx.y;
  const int row0 = blockIdx.x * 16;
  const int t    = threadIdx.x;
  const int w    = t >> 5;              // wave id 0..7
  const int lane = t & 31;
  const int hl   = lane >> 4;           // half-wave: 0 or 1
  const int q    = lane & 15;           // A row / B col / C col within tile
  const int koff = hl * 2;              // K sub-offset per WMMA f32 16x16x4 layout

  // ---- init running state + scan-window tail ----
  for (int i = t; i < 16 * TOPK; i += 256) {
    topvS[i / TOPK][i % TOPK] = NEG_INF;
    topcS[i / TOPK][i % TOPK] = 0x7fffffff;
  }
  for (int i = t; i < 16 * (SW - CHUNK - TOPK); i += 256) {
    int r = i / (SW - CHUNK - TOPK);
    int p = CHUNK + TOPK + i % (SW - CHUNK - TOPK);
    simS[r * SW + p] = NEG_INF;
  }
  if (t < 16) invxS[t] = invx[b * NX + row0 + t];
  __syncthreads();

  // A fragment base: lane q supplies row (row0+q); lanes 16-31 supply K+2,K+3
  const float* Xrow = X + ((size_t)(b * NX + row0 + q)) * DIM + koff;

  for (int c0 = 0; c0 < NY; c0 += CHUNK) {
    // ================= compute phase: 4 tiles of 16x16 per wave =============
#pragma unroll
    for (int ti = 0; ti < CHUNK / (16 * 8); ++ti) {
      int colLocal = w * (CHUNK / 8) + ti * 16;
      int col0     = c0 + colLocal;
      const float* Ycol = Y + ((size_t)(b * NY + col0 + q)) * DIM + koff;
      v8f acc = {};
#pragma unroll 8
      for (int k = 0; k < DIM; k += 4) {
        v2f av = *(const v2f*)(Xrow + k);     // global_load_b64, L1/L2-resident
        v2f bv = *(const v2f*)(Ycol + k);
        acc = __builtin_amdgcn_wmma_f32_16x16x4_f32(
            false, av, false, bv, (short)0, acc, false, false);
      }
      float sy = invy[b * NY + col0 + q] * INV_TAU;
#pragma unroll
      for (int v = 0; v < 8; ++v) {
        int row = v + 8 * hl;                 // C layout: VGPR v -> rows v / v+8
        simS[row * SW + colLocal + q] = acc[v] * invxS[row] * sy;
      }
    }
    __syncthreads();

    // ================= merge phase: wave w owns rows w and w+8 ==============
    for (int rr = 0; rr < 2; ++rr) {
      int r = w + 8 * rr;
      if (lane < TOPK) simS[r * SW + CHUNK + lane] = topvS[r][lane];
      for (int it = 0; it < TOPK; ++it) {
        float bv = NEG_INF;
        int   bc = 0x7fffffff;
        int   bp = 0;
#pragma unroll
        for (int j = 0; j < SW / 32; ++j) {
          int   p = lane + 32 * j;
          float v = simS[r * SW + p];
          int   c = (p < CHUNK) ? (c0 + p)
                                : ((p - CHUNK < TOPK) ? topcS[r][p - CHUNK]
                                                      : 0x7fffffff);
          if (v > bv || (v == bv && c < bc)) { bv = v; bc = c; bp = p; }
        }
#pragma unroll
        for (int off = 16; off; off >>= 1) {    // wave32 (value desc, col asc) max
          float ov = __shfl_xor(bv, off, 32);
          int   oc = __shfl_xor(bc, off, 32);
          int   op = __shfl_xor(bp, off, 32);
          if (ov > bv || (ov == bv && oc < bc)) { bv = ov; bc = oc; bp = op; }
        }
        if (lane == 0) { ntvS[r][it] = bv; ntcS[r][it] = bc; }
        if ((bp & 31) == lane) simS[r * SW + bp] = NEG_INF;  // knock out winner
      }
      if (lane < TOPK) {
        topvS[r][lane] = ntvS[r][lane];
        topcS[r][lane] = ntcS[r][lane];
      }
    }
    __syncthreads();
  }

  // ================= output phase: softmax + sort-by-col + COO ==============
  if (t < 16) {
    int r = t;
    float v[TOPK], e[TOPK];
    int   c[TOPK];
#pragma unroll
    for (int k = 0; k < TOPK; ++k) { v[k] = topvS[r][k]; c[k] = topcS[r][k]; }
    float m = v[0];                       // selected in descending order
    float s = 0.f;
#pragma unroll
    for (int k = 0; k < TOPK; ++k) { e[k] = __expf(v[k] - m); s += e[k]; }
    float inv = 1.0f / s;
    size_t base = ((size_t)(b * NX) + row0 + r) * TOPK;
    // branch-free rank sort by column (columns unique per row)
#pragma unroll
    for (int i = 0; i < TOPK; ++i) {
      int rank = 0;
#pragma unroll
      for (int j = 0; j < TOPK; ++j) rank += (c[j] < c[i]) ? 1 : 0;
      outV[base + rank]                      = e[i] * inv;
      outI[base + rank]                      = b;
      outI[(size_t)TOTN + base + rank]       = row0 + r;
      outI[2 * (size_t)TOTN + base + rank]   = c[i];
    }
  }
}

// ---------------------------------------------------------------------------
extern "C" void kernel_launch(void* const* d_in, const int* in_sizes, int n_in,
                              void* d_out, int out_size, void* d_ws, size_t ws_size,
                              hipStream_t stream) {
  const float* X = (const float*)d_in[0];   // [4, 4096, 256] f32
  const float* Y = (const float*)d_in[1];   // [4, 8192, 256] f32

  float* invx = (float*)d_ws;               // BATCH*NX floats
  float* invy = invx + (size_t)BATCH * NX;  // BATCH*NY floats  (~196 KB total)

  int totalRows = BATCH * (NX + NY);        // 49152 rows, 1 wave each
  int nblocks   = totalRows / 8;            // 8 waves per 256-thread block
  norms_kernel<<<nblocks, 256, 0, stream>>>(X, Y, invx, invy);

  dim3 grid(NX / 16, BATCH);                // 256 x 4 row-tile blocks
  float* outV = (float*)d_out;
  int*   outI = (int*)((float*)d_out + TOTN);
  sim_topk_kernel<<<grid, 256, 0, stream>>>(X, Y, invx, invy, outV, outI);
}